// Encoder_2430951489915
// MI455X (gfx1250) — compile-verified
//
#include <hip/hip_runtime.h>

#define B_DIM 8
#define N_DIM 1024
#define M_DIM 2048
#define OUT_C 128
#define EPSV 1e-8f
#define LOG2E 1.4426950408889634f

typedef __attribute__((ext_vector_type(2))) float v2f;
typedef __attribute__((ext_vector_type(8))) float v8f;

// -------------------------------------------------------------------------
// Stage 1: z[b,m,c] = sum_n feats[b,n,c] * exp(-0.5*((t[b,m]-x[b,n])/s_c)^2)
// then channel-normalize by density. z stored stride-4 (c=0..2, pad) in ws.
// Block = 256 threads = 64 m-values x 4 n-chunks. Grid = (M/64, B).
// Transcendental-bound: 3 v_exp_f32 per (n,m) pair, operands in LDS.
// -------------------------------------------------------------------------
__global__ __launch_bounds__(256) void kde_z_kernel(
    const float* __restrict__ x, const float* __restrict__ y,
    const float* __restrict__ t, const float* __restrict__ sigma,
    float* __restrict__ z)
{
    __shared__ float xs[N_DIM];
    __shared__ float ys0[N_DIM];
    __shared__ float ys1[N_DIM];
    __shared__ float red[4][64][3];

    const int b   = blockIdx.y;
    const int tid = threadIdx.x;

    // stage x, y for this batch into LDS (12 KB)
    for (int i = tid; i < N_DIM; i += 256) {
        xs[i]  = x[b * N_DIM + i];
        ys0[i] = y[(b * N_DIM + i) * 2 + 0];
        ys1[i] = y[(b * N_DIM + i) * 2 + 1];
    }
    __syncthreads();

    // per-channel bandwidth folded with log2(e):  q_c = log2(e) * exp(-2*sigma_c)
    const float q0 = LOG2E * __builtin_amdgcn_exp2f(-2.0f * sigma[0] * LOG2E);
    const float q1 = LOG2E * __builtin_amdgcn_exp2f(-2.0f * sigma[1] * LOG2E);
    const float q2 = LOG2E * __builtin_amdgcn_exp2f(-2.0f * sigma[2] * LOG2E);

    const int m_local = tid & 63;
    const int chunk   = tid >> 6;            // 0..3 -> n sub-range
    const int m       = blockIdx.x * 64 + m_local;
    const float tv    = t[b * M_DIM + m];

    float a0 = 0.0f, a1 = 0.0f, a2 = 0.0f;
    const int n0 = chunk * (N_DIM / 4);
#pragma unroll 4
    for (int n = n0; n < n0 + N_DIM / 4; ++n) {
        const float d = tv - xs[n];
        const float g = -0.5f * d * d;       // shared across channels
        const float e0 = __builtin_amdgcn_exp2f(g * q0);
        const float e1 = __builtin_amdgcn_exp2f(g * q1);
        const float e2 = __builtin_amdgcn_exp2f(g * q2);
        a0 += e0;                            // feats[...,0] == 1
        a1 = fmaf(ys0[n], e1, a1);
        a2 = fmaf(ys1[n], e2, a2);
    }

    red[chunk][m_local][0] = a0;
    red[chunk][m_local][1] = a1;
    red[chunk][m_local][2] = a2;
    __syncthreads();

    if (chunk == 0) {
#pragma unroll
        for (int k = 1; k < 4; ++k) {
            a0 += red[k][m_local][0];
            a1 += red[k][m_local][1];
            a2 += red[k][m_local][2];
        }
        const float density = a0;
        const float inv = 1.0f / (density + EPSV);
        const int row = b * M_DIM + m;
        z[row * 4 + 0] = density;
        z[row * 4 + 1] = a1 * inv;
        z[row * 4 + 2] = a2 * inv;
        z[row * 4 + 3] = 0.0f;
    }
}

// -------------------------------------------------------------------------
// Stage 2: out[row, col] = sum_c z[row,c] * W[col,c] + b_fc[col]
// via V_WMMA_F32_16X16X4_F32 (K padded 3 -> 4 with zeros).
// Block = 256 threads (8 waves). Wave w handles column tile w (8*16 = 128).
// Grid = (B*M)/16 row tiles. Bias preloaded into the C accumulator.
//
// All loads are unconditional (float4 for z, 3 scalars for W); the K-pair
// selection (lanes 0-15: K0/K1, lanes 16-31: K2/K3) is done with register
// selects so no EXEC-divergent path is generated — EXEC stays all-ones
// around the WMMA as the ISA requires.
// -------------------------------------------------------------------------
__global__ __launch_bounds__(256) void proj_wmma_kernel(
    const float* __restrict__ z, const float* __restrict__ W,
    const float* __restrict__ bfc, float* __restrict__ out)
{
    const int tile = blockIdx.x;             // 16-row tile of the 16384 rows
    const int wave = threadIdx.x >> 5;       // column tile 0..7
    const int lane = threadIdx.x & 31;
    const bool hi  = (lane & 16) != 0;       // false: K0/K1 pair, true: K2/K3
    const int l16  = lane & 15;

    // A: 16 rows of z (16B-aligned padded rows -> single global_load_b128)
    const int row = tile * 16 + l16;
    const float4 zr = *(const float4*)(z + row * 4);   // {z0, z1, z2, 0}
    v2f a;
    a.x = hi ? zr.z : zr.x;
    a.y = hi ? zr.w : zr.y;                  // zr.w == 0 (K3 pad)

    // B: 16 columns of W^T, K dim = {W[col,0], W[col,1], W[col,2], 0}
    const int col = wave * 16 + l16;
    const float w0 = W[col * 3 + 0];
    const float w1 = W[col * 3 + 1];
    const float w2 = W[col * 3 + 2];
    v2f bb;
    bb.x = hi ? w2   : w0;
    bb.y = hi ? 0.0f : w1;

    // C: bias broadcast down the 16 rows of each output column
    const float bias = bfc[col];
    v8f c = {bias, bias, bias, bias, bias, bias, bias, bias};

    v8f d = __builtin_amdgcn_wmma_f32_16x16x4_f32(
        /*neg_a=*/false, a, /*neg_b=*/false, bb,
        /*c_mod=*/(short)0, c, /*reuse_a=*/false, /*reuse_b=*/false);

    // D layout: lanes 0-15 -> M=v, lanes 16-31 -> M=8+v; N = lane%16
    const int rbase = tile * 16 + (hi ? 8 : 0);
#pragma unroll
    for (int v = 0; v < 8; ++v) {
        out[(rbase + v) * OUT_C + col] = d[v];
    }
}

extern "C" void kernel_launch(void* const* d_in, const int* in_sizes, int n_in,
                              void* d_out, int out_size, void* d_ws, size_t ws_size,
                              hipStream_t stream) {
    const float* x     = (const float*)d_in[0];   // (8,1024,1)
    const float* y     = (const float*)d_in[1];   // (8,1024,2)
    const float* t     = (const float*)d_in[2];   // (8,2048,1)
    const float* sigma = (const float*)d_in[3];   // (3,)
    const float* W     = (const float*)d_in[4];   // (128,3)
    const float* bfc   = (const float*)d_in[5];   // (128,)
    float* out = (float*)d_out;                   // (8,2048,128)
    float* z   = (float*)d_ws;                    // 16384 * 4 floats = 256 KB

    kde_z_kernel<<<dim3(M_DIM / 64, B_DIM), 256, 0, stream>>>(x, y, t, sigma, z);
    proj_wmma_kernel<<<dim3((B_DIM * M_DIM) / 16), 256, 0, stream>>>(z, W, bfc, out);
}